// JambaMambaMixer_867583393899
// MI455X (gfx1250) — compile-verified
//
#include <hip/hip_runtime.h>
#include <hip/hip_bf16.h>
#include <cstdint>

#define B_   2
#define L_   512
#define H_   2048
#define DI_  4096
#define N_   16
#define K_   4
#define R_   128
#define EPS_ 1e-6f

typedef __attribute__((ext_vector_type(2))) float    v2f;
typedef __attribute__((ext_vector_type(8))) float    v8f;
typedef __attribute__((ext_vector_type(4))) unsigned v4u;
typedef __attribute__((ext_vector_type(4))) int      v4i;
typedef __attribute__((ext_vector_type(8))) int      v8i;

#ifndef __has_builtin
#define __has_builtin(x) 0
#endif
#if __has_builtin(__builtin_amdgcn_tensor_load_to_lds) && \
    __has_builtin(__builtin_amdgcn_s_wait_tensorcnt)
#define USE_TDM 1
#else
#define USE_TDM 0
#endif

static __device__ __forceinline__ v8f wmma_f32_16x16x4(v2f a, v2f b, v8f c) {
  // 8-arg pattern: (neg_a, A, neg_b, B, c_mod, C, reuse_a, reuse_b)
  return __builtin_amdgcn_wmma_f32_16x16x4_f32(false, a, false, b, (short)0, c,
                                               false, false);
}

static __device__ __forceinline__ float sigmoidf_(float x) {
  return 1.0f / (1.0f + __expf(-x));
}
static __device__ __forceinline__ float siluf_(float x) { return x * sigmoidf_(x); }
static __device__ __forceinline__ float softplusf_(float x) {
  return (x > 20.0f) ? x : log1pf(__expf(x));
}

#if USE_TDM
// ---------------------------------------------------------------------------
// Issue one TDM DMA: load a (tile_rows x BKE) fp32 tile from a row-major
// matrix (row stride = stride_elems) into LDS at lds_byte_addr, with hardware
// padding of +4 DWORDs after every 16 DWORDs (so LDS row stride = BKE+4 when
// BKE==16). Descriptor layout per CDNA5 ISA.
// Tracked on TENSORcnt; EXEC is ignored (call from one wave only).
// ---------------------------------------------------------------------------
template<int BKE>
static __device__ __forceinline__ void tdm_load_tile_f32(
    const float* gsrc, unsigned lds_byte_addr, int tile_rows, int stride_elems)
{
  static_assert(BKE == 16, "pad config assumes 16-dword rows");
  const unsigned long long ga = (unsigned long long)(uintptr_t)gsrc;
  // Group 0: count=1 | lds_addr | global_addr(57b) | type=2
  v4u g0 = { 1u,
             lds_byte_addr,
             (unsigned)(ga & 0xFFFFFFFFu),
             (unsigned)((ga >> 32) & 0x01FFFFFFu) | (2u << 30) };
  // Group 1: mask=0, data_size=4B(code 2), pad_enable=1,
  //          pad_interval=code 3 (16 DWORDs), pad_amount=code 3 (4 DWORDs)
  const unsigned d0  = (2u << 16) | (1u << 20) | (3u << 22) | (3u << 25);
  const unsigned td0 = 0x40000000u;              // huge bounds: no OOB clip
  const unsigned td1 = 0x40000000u;
  const unsigned s0  = (unsigned)stride_elems;   // dim0 stride (elements)
  v8i g1 = { (int)d0,
             (int)((td0 & 0xFFFFu) << 16),
             (int)((td0 >> 16) | ((td1 & 0xFFFFu) << 16)),
             (int)((td1 >> 16) | ((unsigned)BKE << 16)),   // tile_dim0 = BKE
             (int)(unsigned)tile_rows,                     // tile_dim1
             (int)s0, 0, 0 };                              // stride lo32/hi16
  v4i g2 = { 0, 0, 0, 0 };                                 // 2D: unused
  v4i g3 = { 0, 0, 0, 0 };
#if __clang_major__ >= 23
  v8i gx = { 0, 0, 0, 0, 0, 0, 0, 0 };
  __builtin_amdgcn_tensor_load_to_lds(g0, g1, g2, g3, gx, 0);
#else
  __builtin_amdgcn_tensor_load_to_lds(g0, g1, g2, g3, 0);
#endif
}
#endif  // USE_TDM

// ---------------------------------------------------------------------------
// C[M,N] = A[M,K] * W[N,K]^T   (all fp32 row-major), via V_WMMA_F32_16X16X4_F32
// with TDM-driven double-buffered LDS tiles. EPI==1: C = softplus(C + bias[n]).
// Requires M % BM == 0, N % BN == 0, Kd % BK == 0.
// ---------------------------------------------------------------------------
template<int BM, int BN, int BK, int WAVES_M, int WAVES_N, int EPI>
__global__ __launch_bounds__(WAVES_M * WAVES_N * 32)
void gemm_wmma_f32(const float* __restrict__ A, const float* __restrict__ W,
                   const float* __restrict__ bias, float* __restrict__ C,
                   int M, int N, int Kd)
{
  constexpr int NTHREADS = WAVES_M * WAVES_N * 32;
  constexpr int WTM = BM / WAVES_M;     // rows per wave
  constexpr int WTN = BN / WAVES_N;     // cols per wave
  constexpr int TM  = WTM / 16;
  constexpr int TN  = WTN / 16;
  constexpr int LDSS = BK + 4;          // padded LDS row stride (TDM pad)

  __shared__ float sA[2][BM][LDSS];
  __shared__ float sW[2][BN][LDSS];

  const int tid  = threadIdx.x;
  const int wave = tid >> 5;
  const int lane = tid & 31;
  const int wm   = wave / WAVES_N;
  const int wn   = wave % WAVES_N;
  const int hh   = lane >> 4;           // lane half (0/1)
  const int lrow = lane & 15;

  const int block_m = blockIdx.y * BM;
  const int block_n = blockIdx.x * BN;

  v8f acc[TM][TN];
#pragma unroll
  for (int i = 0; i < TM; ++i)
#pragma unroll
    for (int j = 0; j < TN; ++j) {
      v8f z = {0.f, 0.f, 0.f, 0.f, 0.f, 0.f, 0.f, 0.f};
      acc[i][j] = z;
    }

  const int nk = Kd / BK;

#if USE_TDM
  if (tid < 32) {  // wave 0 drives the Tensor Data Mover
    tdm_load_tile_f32<BK>(&A[(size_t)block_m * Kd], (unsigned)(uintptr_t)&sA[0][0][0], BM, Kd);
    tdm_load_tile_f32<BK>(&W[(size_t)block_n * Kd], (unsigned)(uintptr_t)&sW[0][0][0], BN, Kd);
  }
#endif

  for (int it = 0; it < nk; ++it) {
    const int cur = it & 1;
#if USE_TDM
    if (tid < 32) {
      if (it + 1 < nk) {
        const int k1 = (it + 1) * BK;
        tdm_load_tile_f32<BK>(&A[(size_t)block_m * Kd + k1],
                              (unsigned)(uintptr_t)&sA[cur ^ 1][0][0], BM, Kd);
        tdm_load_tile_f32<BK>(&W[(size_t)block_n * Kd + k1],
                              (unsigned)(uintptr_t)&sW[cur ^ 1][0][0], BN, Kd);
        __builtin_amdgcn_s_wait_tensorcnt(2);  // tile `it` has landed
      } else {
        __builtin_amdgcn_s_wait_tensorcnt(0);  // last tile has landed
      }
    }
#else
    {
      const int k0 = it * BK;
      for (int i = tid; i < BM * (BK / 4); i += NTHREADS) {
        const int r  = i / (BK / 4);
        const int c4 = i % (BK / 4);
        const float4 v = *reinterpret_cast<const float4*>(
            &A[(size_t)(block_m + r) * Kd + k0 + c4 * 4]);
        *reinterpret_cast<float4*>(&sA[cur][r][c4 * 4]) = v;
      }
      for (int i = tid; i < BN * (BK / 4); i += NTHREADS) {
        const int r  = i / (BK / 4);
        const int c4 = i % (BK / 4);
        const float4 v = *reinterpret_cast<const float4*>(
            &W[(size_t)(block_n + r) * Kd + k0 + c4 * 4]);
        *reinterpret_cast<float4*>(&sW[cur][r][c4 * 4]) = v;
      }
    }
#endif
    __syncthreads();   // tile `it` visible to all waves

#pragma unroll
    for (int kk = 0; kk < BK; kk += 4) {
      v2f afrag[TM];
      v2f bfrag[TN];
      // 16x4 A fragment: lanes 0-15 rows M, VGPR0 = K{0|2}, VGPR1 = K{1|3}
#pragma unroll
      for (int i = 0; i < TM; ++i) {
        const int m = wm * WTM + i * 16 + lrow;
        afrag[i][0] = sA[cur][m][kk + 2 * hh];
        afrag[i][1] = sA[cur][m][kk + 2 * hh + 1];
      }
      // 4x16 B fragment = W^T: lane = N column, mirrored K striping
#pragma unroll
      for (int j = 0; j < TN; ++j) {
        const int n = wn * WTN + j * 16 + lrow;
        bfrag[j][0] = sW[cur][n][kk + 2 * hh];
        bfrag[j][1] = sW[cur][n][kk + 2 * hh + 1];
      }
#pragma unroll
      for (int i = 0; i < TM; ++i)
#pragma unroll
        for (int j = 0; j < TN; ++j)
          acc[i][j] = wmma_f32_16x16x4(afrag[i], bfrag[j], acc[i][j]);
    }
    __syncthreads();   // all waves done with buffer `cur` before TDM reuses it
  }

  // C/D layout: VGPR r -> row (r + 8*half), lane%16 -> column
#pragma unroll
  for (int i = 0; i < TM; ++i)
#pragma unroll
    for (int j = 0; j < TN; ++j) {
      const int ncol = block_n + wn * WTN + j * 16 + lrow;
      float bv = 0.f;
      if constexpr (EPI == 1) bv = bias[ncol];
#pragma unroll
      for (int r = 0; r < 8; ++r) {
        const int mrow = block_m + wm * WTM + i * 16 + r + 8 * hh;
        float v = acc[i][j][r];
        if constexpr (EPI == 1) v = softplusf_(v + bv);
        C[(size_t)mrow * N + ncol] = v;
      }
    }
}

// ---------------------------------------------------------------------------
// Depthwise causal conv1d (K=4) + bias + SiLU. proj=[B,L,2*DI] (first DI used),
// output hconv=[B,L,DI]. One thread per channel, 64-timestep chunks.
// ---------------------------------------------------------------------------
__global__ __launch_bounds__(256)
void conv_silu_kernel(const float* __restrict__ proj,
                      const float* __restrict__ cw,
                      const float* __restrict__ cb,
                      float* __restrict__ hconv)
{
  const int d  = blockIdx.x * 256 + threadIdx.x;
  const int t0 = blockIdx.y * 64;
  const int b  = blockIdx.z;
  const float w0 = cw[d * 4 + 0], w1 = cw[d * 4 + 1];
  const float w2 = cw[d * 4 + 2], w3 = cw[d * 4 + 3];
  const float bb = cb[d];
  const size_t rs = (size_t)(2 * DI_);
  const float* base  = proj + (size_t)b * L_ * rs + d;
  float* obase = hconv + (size_t)b * L_ * DI_ + d;
  float xm3 = (t0 >= 3) ? base[(size_t)(t0 - 3) * rs] : 0.f;
  float xm2 = (t0 >= 2) ? base[(size_t)(t0 - 2) * rs] : 0.f;
  float xm1 = (t0 >= 1) ? base[(size_t)(t0 - 1) * rs] : 0.f;
  for (int t = t0; t < t0 + 64; ++t) {
    const float xc = base[(size_t)t * rs];
    const float s  = w0 * xm3 + w1 * xm2 + w2 * xm1 + w3 * xc + bb;
    obase[(size_t)t * DI_] = siluf_(s);
    xm3 = xm2; xm2 = xm1; xm1 = xc;
  }
}

// ---------------------------------------------------------------------------
// Split RMSNorm over ssm rows [BL, R+2N]: dt (128, full-wave reduce),
// B and C (16 each, half-wave reduce; lanes 0-15 -> B, 16-31 -> C).
// One wave32 per row, 8 rows per block.
// ---------------------------------------------------------------------------
__global__ __launch_bounds__(256)
void rmsnorm_split_kernel(const float* __restrict__ ssm,
                          const float* __restrict__ dt_ln,
                          const float* __restrict__ b_ln,
                          const float* __restrict__ c_ln,
                          float* __restrict__ dtn,
                          float* __restrict__ Bn,
                          float* __restrict__ Cn)
{
  const int wave = threadIdx.x >> 5;
  const int lane = threadIdx.x & 31;
  const int row  = blockIdx.x * 8 + wave;
  const float* srow = ssm + (size_t)row * (R_ + 2 * N_);

  // dt part (128 wide)
  float4 dv = *reinterpret_cast<const float4*>(&srow[lane * 4]);
  float ss = dv.x * dv.x + dv.y * dv.y + dv.z * dv.z + dv.w * dv.w;
  for (int m = 16; m >= 1; m >>= 1) ss += __shfl_xor(ss, m, 32);
  const float sc = rsqrtf(ss * (1.0f / R_) + EPS_);
  const float4 wv = *reinterpret_cast<const float4*>(&dt_ln[lane * 4]);
  float4 ov;
  ov.x = dv.x * sc * wv.x; ov.y = dv.y * sc * wv.y;
  ov.z = dv.z * sc * wv.z; ov.w = dv.w * sc * wv.w;
  *reinterpret_cast<float4*>(&dtn[(size_t)row * R_ + lane * 4]) = ov;

  // B (lanes 0-15) and C (lanes 16-31), each 16 wide
  const float v = srow[R_ + lane];
  float s2 = v * v;
  for (int m = 8; m >= 1; m >>= 1) s2 += __shfl_xor(s2, m, 32);
  const float sc2 = rsqrtf(s2 * (1.0f / N_) + EPS_);
  const float w2  = (lane < N_) ? b_ln[lane] : c_ln[lane - N_];
  const float o2  = v * sc2 * w2;
  if (lane < N_) Bn[(size_t)row * N_ + lane] = o2;
  else           Cn[(size_t)row * N_ + (lane - N_)] = o2;
}

// ---------------------------------------------------------------------------
// Selective scan: one thread per (b, d). 16-wide state in registers, B/C
// staged in LDS per 64-step chunk. Fused D-skip connection + SiLU gate.
// ---------------------------------------------------------------------------
__global__ __launch_bounds__(256)
void selective_scan_kernel(const float* __restrict__ dt,
                           const float* __restrict__ hconv,
                           const float* __restrict__ Bn,
                           const float* __restrict__ Cn,
                           const float* __restrict__ A_log,
                           const float* __restrict__ Dp,
                           const float* __restrict__ proj,
                           float* __restrict__ y)
{
  constexpr int TS = 64;
  __shared__ float sB[TS][N_];
  __shared__ float sC[TS][N_];
  const int d = blockIdx.x * 256 + threadIdx.x;
  const int b = blockIdx.y;

  float Arow[N_];
#pragma unroll
  for (int n = 0; n < N_; ++n) Arow[n] = -__expf(A_log[(size_t)d * N_ + n]);
  const float Dv = Dp[d];
  float st[N_];
#pragma unroll
  for (int n = 0; n < N_; ++n) st[n] = 0.f;

  const size_t rowbase = (size_t)b * L_;
  for (int t0 = 0; t0 < L_; t0 += TS) {
    for (int i = threadIdx.x; i < TS * N_; i += 256) {
      sB[i / N_][i % N_] = Bn[(rowbase + t0) * N_ + i];
      sC[i / N_][i % N_] = Cn[(rowbase + t0) * N_ + i];
    }
    __syncthreads();
    for (int tt = 0; tt < TS; ++tt) {
      const size_t t   = rowbase + t0 + tt;
      const size_t idx = t * DI_ + d;
      const float dtv = dt[idx];
      const float hv  = hconv[idx];
      const float gv  = proj[t * (2 * DI_) + DI_ + d];
      const float dh  = dtv * hv;
      float acc = 0.f;
#pragma unroll
      for (int n = 0; n < N_; ++n) {
        const float dA = __expf(dtv * Arow[n]);
        st[n] = st[n] * dA + dh * sB[tt][n];
        acc += st[n] * sC[tt][n];
      }
      y[idx] = (acc + hv * Dv) * siluf_(gv);
    }
    __syncthreads();
  }
}

// ---------------------------------------------------------------------------
extern "C" void kernel_launch(void* const* d_in, const int* in_sizes, int n_in,
                              void* d_out, int out_size, void* d_ws, size_t ws_size,
                              hipStream_t stream) {
  (void)in_sizes; (void)n_in; (void)out_size; (void)ws_size;
  constexpr int BL = B_ * L_;  // 1024 token rows

  const float* x      = (const float*)d_in[0];   // [B,L,H]
  const float* w_in   = (const float*)d_in[1];   // [2*DI, H]
  const float* conv_w = (const float*)d_in[2];   // [DI, K]
  const float* conv_b = (const float*)d_in[3];   // [DI]
  const float* w_x    = (const float*)d_in[4];   // [R+2N, DI]
  const float* dt_ln  = (const float*)d_in[5];   // [R]
  const float* b_ln   = (const float*)d_in[6];   // [N]
  const float* c_ln   = (const float*)d_in[7];   // [N]
  const float* w_dt   = (const float*)d_in[8];   // [DI, R]
  const float* b_dt   = (const float*)d_in[9];   // [DI]
  const float* A_log  = (const float*)d_in[10];  // [DI, N]
  const float* Dp     = (const float*)d_in[11];  // [DI]
  const float* w_out  = (const float*)d_in[12];  // [H, DI]
  float* out = (float*)d_out;                    // [B,L,H] fp32

  float* ws = (float*)d_ws;
  float* proj  = ws; ws += (size_t)BL * 2 * DI_;       // [BL, 2*DI]
  float* hconv = ws; ws += (size_t)BL * DI_;           // [BL, DI]
  float* ssm   = ws; ws += (size_t)BL * (R_ + 2 * N_); // [BL, 160]
  float* dtn   = ws; ws += (size_t)BL * R_;            // [BL, 128]
  float* Bnrm  = ws; ws += (size_t)BL * N_;            // [BL, 16]
  float* Cnrm  = ws; ws += (size_t)BL * N_;            // [BL, 16]
  float* dtv   = ws; ws += (size_t)BL * DI_;           // [BL, DI]
  float* yv    = ws; ws += (size_t)BL * DI_;           // [BL, DI]

  // 1) in_proj: [1024,2048] x [8192,2048]^T -> [1024,8192]
  gemm_wmma_f32<128, 128, 16, 4, 2, 0>
      <<<dim3(2 * DI_ / 128, BL / 128), 256, 0, stream>>>(
          x, w_in, nullptr, proj, BL, 2 * DI_, H_);

  // 2) depthwise causal conv + SiLU
  conv_silu_kernel<<<dim3(DI_ / 256, L_ / 64, B_), 256, 0, stream>>>(
      proj, conv_w, conv_b, hconv);

  // 3) x_proj: [1024,4096] x [160,4096]^T -> [1024,160]
  gemm_wmma_f32<128, 32, 16, 4, 2, 0>
      <<<dim3((R_ + 2 * N_) / 32, BL / 128), 256, 0, stream>>>(
          hconv, w_x, nullptr, ssm, BL, R_ + 2 * N_, DI_);

  // 4) split RMSNorm of dt / B / C
  rmsnorm_split_kernel<<<BL / 8, 256, 0, stream>>>(
      ssm, dt_ln, b_ln, c_ln, dtn, Bnrm, Cnrm);

  // 5) dt_proj + bias + softplus: [1024,128] x [4096,128]^T -> [1024,4096]
  gemm_wmma_f32<128, 128, 16, 4, 2, 1>
      <<<dim3(DI_ / 128, BL / 128), 256, 0, stream>>>(
          dtn, w_dt, b_dt, dtv, BL, DI_, R_);

  // 6) selective scan + D skip + SiLU gate
  selective_scan_kernel<<<dim3(DI_ / 256, B_), 256, 0, stream>>>(
      dtv, hconv, Bnrm, Cnrm, A_log, Dp, proj, yv);

  // 7) out_proj: [1024,4096] x [2048,4096]^T -> [1024,2048]
  gemm_wmma_f32<128, 128, 16, 4, 2, 0>
      <<<dim3(H_ / 128, BL / 128), 256, 0, stream>>>(
          yv, w_out, nullptr, out, BL, H_, DI_);
}